// GAT_40200893890740
// MI455X (gfx1250) — compile-verified
//
#include <hip/hip_runtime.h>
#include <math.h>

#define N_NODES 50000
#define N_EDGES 800000
#define ETOT    (N_EDGES + N_NODES)
#define F_IN    128
#define HID     64
#define HEADS   2
#define OUT_F   40
#define EDIM    5
#define NEG_SLOPE 0.2f

#define CDIV(a,b) (((a)+(b)-1)/(b))

typedef __attribute__((ext_vector_type(2))) float v2f;
typedef __attribute__((ext_vector_type(8))) float v8f;

// ---------- monotonic float<->uint encoding for atomic max ----------
__device__ inline unsigned encf(float f) {
    unsigned u = __float_as_uint(f);
    return (u & 0x80000000u) ? ~u : (u | 0x80000000u);
}
__device__ inline float decf(unsigned m) {
    unsigned u = (m & 0x80000000u) ? (m ^ 0x80000000u) : ~m;
    return __uint_as_float(u);
}
#define ENC_NEG_INF 0x007FFFFFu  // encf(-inf)

// ---------- fills ----------
__global__ void fill_f32(float* p, int n, float v) {
    int i = blockIdx.x * blockDim.x + threadIdx.x;
    if (i < n) p[i] = v;
}
__global__ void fill_u32(unsigned* p, int n, unsigned v) {
    int i = blockIdx.x * blockDim.x + threadIdx.x;
    if (i < n) p[i] = v;
}

// ---------- zero-pad W2 [K,N] -> Wp [K,NPAD] ----------
__global__ void pad_b(const float* __restrict__ W, float* __restrict__ Wp,
                      int K, int N, int NPAD) {
    int i = blockIdx.x * blockDim.x + threadIdx.x;
    if (i >= K * NPAD) return;
    int r = i / NPAD, c = i % NPAD;
    Wp[i] = (c < N) ? W[r * N + c] : 0.0f;
}

// ---------- self-loop attr construction (fill_value='mean') ----------
__global__ void selfloop_accum(const int* __restrict__ dst, const float* __restrict__ eattr,
                               float* cnt, float* loop5, int E) {
    int e = blockIdx.x * blockDim.x + threadIdx.x;
    if (e >= E) return;
    int d = dst[e];
    atomicAdd(&cnt[d], 1.0f);
#pragma unroll
    for (int i = 0; i < EDIM; i++)
        atomicAdd(&loop5[d * EDIM + i], eattr[(size_t)e * EDIM + i]);
}
__global__ void selfloop_final(float* loop5, const float* cnt, int n) {
    int i = blockIdx.x * blockDim.x + threadIdx.x;
    if (i >= n * EDIM) return;
    float c = fmaxf(cnt[i / EDIM], 1.0f);
    loop5[i] /= c;
}

// ---------- fold We @ att_e into a tiny [EDIM,H] vector ----------
__global__ void precompute_v(const float* We1, const float* ae1,
                             const float* We2, const float* ae2,
                             float* v1, float* v2) {
    int t = threadIdx.x;
    if (t < EDIM * HEADS) {            // v1[d*2+h]
        int d = t / HEADS, h = t % HEADS;
        float s = 0.f;
        for (int c = 0; c < HID; c++)
            s += We1[d * (HEADS * HID) + h * HID + c] * ae1[h * HID + c];
        v1[t] = s;
    } else if (t < EDIM * HEADS + EDIM) {  // v2[d]
        int d = t - EDIM * HEADS;
        float s = 0.f;
        for (int c = 0; c < OUT_F; c++)
            s += We2[d * OUT_F + c] * ae2[c];
        v2[d] = s;
    }
}

// ---------- WMMA f32 GEMM: C[M,N] = A[M,K] @ B[K,NPAD], M % 16 == 0 ----------
// B row stride is NPAD = NT*16 (zero-padded beyond N); loads are branchless.
// One wave: a 16-row stripe and all NT column tiles; K in steps of 4.
template<int NT, int K>
__global__ void wmma_gemm(const float* __restrict__ A, const float* __restrict__ B,
                          float* __restrict__ C, int M, int N) {
    constexpr int NPAD = NT * 16;
    int wave = (blockIdx.x * blockDim.x + threadIdx.x) >> 5;
    int lane = threadIdx.x & 31;
    int half = lane >> 4;      // 0: lanes 0-15, 1: lanes 16-31
    int lrow = lane & 15;
    int mtiles = M >> 4;
    if (wave >= mtiles) return;
    int m0 = wave << 4;

    v8f acc[NT];
#pragma unroll
    for (int nt = 0; nt < NT; nt++)
#pragma unroll
        for (int g = 0; g < 8; g++) acc[nt][g] = 0.0f;

    const float* arow = A + (size_t)(m0 + lrow) * K;
    const float* bcol = B + lrow;            // per-lane column base
#pragma unroll 4
    for (int k = 0; k < K; k += 4) {
        int kk = k + half * 2;               // A/B K-slot: half*2 + vgpr
        v2f a;
        a.x = arow[kk];
        a.y = arow[kk + 1];
#pragma unroll
        for (int nt = 0; nt < NT; nt++) {
            v2f b;
            b.x = bcol[kk * NPAD + nt * 16];
            b.y = bcol[(kk + 1) * NPAD + nt * 16];
            acc[nt] = __builtin_amdgcn_wmma_f32_16x16x4_f32(
                false, a, false, b, (short)0, acc[nt], false, false);
        }
    }
#pragma unroll
    for (int nt = 0; nt < NT; nt++) {
        int col = nt * 16 + lrow;
        if (col < N) {
#pragma unroll
            for (int g = 0; g < 8; g++)      // C row = vgpr + half*8
                C[(size_t)(m0 + g + half * 8) * N + col] = acc[nt][g];
        }
    }
}

// ---------- per-node attention scores: als/ald = <xh[n,h,:], att> ----------
__global__ void node_scores(const float* __restrict__ xh,
                            const float* __restrict__ asrc, const float* __restrict__ adst,
                            float* als, float* ald, int n, int H, int C) {
    int i = blockIdx.x * blockDim.x + threadIdx.x;  // i = node*H + h
    if (i >= n * H) return;
    int node = i / H, h = i % H;
    const float* row = xh + (size_t)node * H * C + h * C;
    float s = 0.f, d = 0.f;
    for (int c = 0; c < C; c++) {
        float v = row[c];
        s += v * asrc[h * C + c];
        d += v * adst[h * C + c];
    }
    als[i] = s;
    ald[i] = d;
}

// ---------- edge pass 1: raw alpha + segment max over dst ----------
__global__ void edge_pass1(const int* __restrict__ src, const int* __restrict__ dst,
                           const float* __restrict__ eattr, const float* __restrict__ loop5,
                           const float* __restrict__ als, const float* __restrict__ ald,
                           const float* __restrict__ v,
                           float* alpha, unsigned* amax, int E, int Etot, int H) {
    int e = blockIdx.x * blockDim.x + threadIdx.x;
    if (e >= Etot) return;
    int s, d;
    float ea[EDIM];
    if (e < E) {
        s = src[e]; d = dst[e];
#pragma unroll
        for (int i = 0; i < EDIM; i++) ea[i] = eattr[(size_t)e * EDIM + i];
    } else {
        s = d = e - E;
#pragma unroll
        for (int i = 0; i < EDIM; i++) ea[i] = loop5[s * EDIM + i];
    }
    for (int h = 0; h < H; h++) {
        float ale = 0.f;
#pragma unroll
        for (int i = 0; i < EDIM; i++) ale += ea[i] * v[i * H + h];
        float a = als[s * H + h] + ald[d * H + h] + ale;
        a = (a > 0.f) ? a : NEG_SLOPE * a;       // leaky relu
        alpha[(size_t)e * H + h] = a;
        atomicMax(&amax[d * H + h], encf(a));
    }
}

// ---------- edge pass 2: exp(alpha - max) + segment sum ----------
__global__ void edge_pass2(const int* __restrict__ dst, float* alpha,
                           const unsigned* __restrict__ amax, float* denom,
                           int E, int Etot, int H) {
    int e = blockIdx.x * blockDim.x + threadIdx.x;
    if (e >= Etot) return;
    int d = (e < E) ? dst[e] : (e - E);
    for (int h = 0; h < H; h++) {
        float a = alpha[(size_t)e * H + h];
        float ex = expf(a - decf(amax[d * H + h]));
        alpha[(size_t)e * H + h] = ex;
        atomicAdd(&denom[d * H + h], ex);
    }
}

// ---------- edge pass 3: weighted scatter-add (thread per edge-channel) ----------
__global__ void edge_aggregate(const int* __restrict__ src, const int* __restrict__ dst,
                               const float* __restrict__ alpha, const float* __restrict__ denom,
                               const float* __restrict__ xh, float* acc,
                               int E, int Etot, int H, int C) {
    int idx = blockIdx.x * blockDim.x + threadIdx.x;
    int HC = H * C;
    if (idx >= Etot * HC) return;
    int e = idx / HC, c = idx % HC, h = c / C;
    int s = (e < E) ? src[e] : (e - E);
    int d = (e < E) ? dst[e] : (e - E);
    float w = alpha[(size_t)e * H + h] / (denom[d * H + h] + 1e-16f);
    atomicAdd(&acc[(size_t)d * HC + c], xh[(size_t)s * HC + c] * w);
}

// ---------- bias + ELU (in place) ----------
__global__ void bias_elu(float* acc, const float* __restrict__ b, int total, int F) {
    int i = blockIdx.x * blockDim.x + threadIdx.x;
    if (i >= total) return;
    float x = acc[i] + b[i % F];
    acc[i] = (x > 0.f) ? x : expm1f(x);
}

// ---------- bias + log_softmax over OUT_F ----------
__global__ void bias_logsoftmax(const float* __restrict__ acc, const float* __restrict__ b,
                                float* __restrict__ out, int n) {
    int node = blockIdx.x * blockDim.x + threadIdx.x;
    if (node >= n) return;
    float vals[OUT_F];
    float mx = -INFINITY;
#pragma unroll
    for (int c = 0; c < OUT_F; c++) {
        float v = acc[(size_t)node * OUT_F + c] + b[c];
        vals[c] = v;
        mx = fmaxf(mx, v);
    }
    float sum = 0.f;
#pragma unroll
    for (int c = 0; c < OUT_F; c++) sum += expf(vals[c] - mx);
    float lse = mx + logf(sum);
#pragma unroll
    for (int c = 0; c < OUT_F; c++) out[(size_t)node * OUT_F + c] = vals[c] - lse;
}

extern "C" void kernel_launch(void* const* d_in, const int* in_sizes, int n_in,
                              void* d_out, int out_size, void* d_ws, size_t ws_size,
                              hipStream_t stream) {
    const float* x   = (const float*)d_in[0];
    const int*   ei  = (const int*)d_in[1];
    const float* ea  = (const float*)d_in[2];
    const float* W1  = (const float*)d_in[3];
    const float* as1 = (const float*)d_in[4];
    const float* ad1 = (const float*)d_in[5];
    const float* We1 = (const float*)d_in[6];
    const float* ae1 = (const float*)d_in[7];
    const float* b1  = (const float*)d_in[8];
    const float* W2  = (const float*)d_in[9];
    const float* as2 = (const float*)d_in[10];
    const float* ad2 = (const float*)d_in[11];
    const float* We2 = (const float*)d_in[12];
    const float* ae2 = (const float*)d_in[13];
    const float* b2  = (const float*)d_in[14];

    const int N = N_NODES, E = N_EDGES, Et = ETOT;
    const int HC1 = HEADS * HID;  // 128
    const int NPAD2 = 48;         // OUT_F padded to 3 tiles of 16
    const int* src = ei;
    const int* dst = ei + E;

    // ---- carve workspace ----
    char* ws = (char*)d_ws;
    size_t off = 0;
    auto alloc = [&](size_t bytes) -> void* {
        void* p = ws + off;
        off = (off + bytes + 255) & ~(size_t)255;
        return p;
    };
    float*    cnt    = (float*)alloc(sizeof(float) * N);
    float*    loop5  = (float*)alloc(sizeof(float) * N * EDIM);
    float*    v1     = (float*)alloc(sizeof(float) * 16);
    float*    v2     = (float*)alloc(sizeof(float) * 16);
    float*    W2p    = (float*)alloc(sizeof(float) * F_IN * NPAD2);      // padded W2
    float*    xh1    = (float*)alloc(sizeof(float) * (size_t)N * HC1);
    float*    als1   = (float*)alloc(sizeof(float) * N * HEADS);
    float*    ald1   = (float*)alloc(sizeof(float) * N * HEADS);
    unsigned* amax1  = (unsigned*)alloc(sizeof(unsigned) * N * HEADS);
    float*    den1   = (float*)alloc(sizeof(float) * N * HEADS);
    float*    alpha1 = (float*)alloc(sizeof(float) * (size_t)Et * HEADS);
    float*    acc1   = (float*)alloc(sizeof(float) * (size_t)N * HC1);   // becomes h after bias+elu
    float*    xh2    = (float*)alloc(sizeof(float) * (size_t)N * OUT_F);
    float*    als2   = (float*)alloc(sizeof(float) * N);
    float*    ald2   = (float*)alloc(sizeof(float) * N);
    unsigned* amax2  = (unsigned*)alloc(sizeof(unsigned) * N);
    float*    den2   = (float*)alloc(sizeof(float) * N);
    float*    alpha2 = (float*)alloc(sizeof(float) * Et);
    float*    acc2   = (float*)alloc(sizeof(float) * (size_t)N * OUT_F);
    (void)ws_size; (void)n_in; (void)in_sizes; (void)out_size;

    const int T = 256;

    // ---- self-loop attrs (mean of incoming) ----
    fill_f32<<<CDIV(N, T), T, 0, stream>>>(cnt, N, 0.f);
    fill_f32<<<CDIV(N * EDIM, T), T, 0, stream>>>(loop5, N * EDIM, 0.f);
    selfloop_accum<<<CDIV(E, T), T, 0, stream>>>(dst, ea, cnt, loop5, E);
    selfloop_final<<<CDIV(N * EDIM, T), T, 0, stream>>>(loop5, cnt, N);
    precompute_v<<<1, 32, 0, stream>>>(We1, ae1, We2, ae2, v1, v2);
    pad_b<<<CDIV(F_IN * NPAD2, T), T, 0, stream>>>(W2, W2p, F_IN, OUT_F, NPAD2);

    // ---- layer 1: xh1 = x @ W1 (N already == NPAD, W1 used directly) ----
    {
        int mtiles = N / 16, waves_per_blk = 128 / 32;
        wmma_gemm<8, F_IN><<<CDIV(mtiles, waves_per_blk), 128, 0, stream>>>(x, W1, xh1, N, HC1);
    }
    node_scores<<<CDIV(N * HEADS, T), T, 0, stream>>>(xh1, as1, ad1, als1, ald1, N, HEADS, HID);
    fill_u32<<<CDIV(N * HEADS, T), T, 0, stream>>>(amax1, N * HEADS, ENC_NEG_INF);
    fill_f32<<<CDIV(N * HEADS, T), T, 0, stream>>>(den1, N * HEADS, 0.f);
    fill_f32<<<CDIV(N * HC1, T), T, 0, stream>>>(acc1, N * HC1, 0.f);
    edge_pass1<<<CDIV(Et, T), T, 0, stream>>>(src, dst, ea, loop5, als1, ald1, v1,
                                              alpha1, amax1, E, Et, HEADS);
    edge_pass2<<<CDIV(Et, T), T, 0, stream>>>(dst, alpha1, amax1, den1, E, Et, HEADS);
    edge_aggregate<<<CDIV(Et * HC1, T), T, 0, stream>>>(src, dst, alpha1, den1, xh1, acc1,
                                                        E, Et, HEADS, HID);
    bias_elu<<<CDIV(N * HC1, T), T, 0, stream>>>(acc1, b1, N * HC1, HC1);

    // ---- layer 2: xh2 = h @ W2 (padded B, branchless loads) ----
    {
        int mtiles = N / 16, waves_per_blk = 128 / 32;
        wmma_gemm<3, HC1><<<CDIV(mtiles, waves_per_blk), 128, 0, stream>>>(acc1, W2p, xh2, N, OUT_F);
    }
    node_scores<<<CDIV(N, T), T, 0, stream>>>(xh2, as2, ad2, als2, ald2, N, 1, OUT_F);
    fill_u32<<<CDIV(N, T), T, 0, stream>>>(amax2, N, ENC_NEG_INF);
    fill_f32<<<CDIV(N, T), T, 0, stream>>>(den2, N, 0.f);
    fill_f32<<<CDIV(N * OUT_F, T), T, 0, stream>>>(acc2, N * OUT_F, 0.f);
    edge_pass1<<<CDIV(Et, T), T, 0, stream>>>(src, dst, ea, loop5, als2, ald2, v2,
                                              alpha2, amax2, E, Et, 1);
    edge_pass2<<<CDIV(Et, T), T, 0, stream>>>(dst, alpha2, amax2, den2, E, Et, 1);
    edge_aggregate<<<CDIV(Et * OUT_F, T), T, 0, stream>>>(src, dst, alpha2, den2, xh2, acc2,
                                                          E, Et, 1, OUT_F);
    bias_logsoftmax<<<CDIV(N, T), T, 0, stream>>>(acc2, b2, (float*)d_out, N);
}